// completeNet_15556371546909
// MI455X (gfx1250) — compile-verified
//
#include <hip/hip_runtime.h>
#include <math.h>

// ---------------------------------------------------------------------------
// CDNA5 (gfx1250) f32 WMMA:  V_WMMA_F32_16X16X4_F32,  D(16x16) = A(16x4) x B(4x16) + C
// A per lane: 2 floats (lanes 0-15: K=k0,k0+1 ; lanes 16-31: K=k0+2,k0+3), row = lane&15
// B per lane: 2 floats (K=k0+2*half, k0+2*half+1), col = lane&15
// C/D: 8 VGPRs, row = r + 8*(lane>>4), col = lane&15
// ---------------------------------------------------------------------------
typedef float v2f __attribute__((ext_vector_type(2)));
typedef float v8f __attribute__((ext_vector_type(8)));

__device__ __forceinline__ v8f wmma_f32(v2f a, v2f b, v8f c) {
    return __builtin_amdgcn_wmma_f32_16x16x4_f32(
        /*neg_a=*/false, a, /*neg_b=*/false, b,
        /*c_mod=*/(short)0, c, /*reuse_a=*/false, /*reuse_b=*/false);
}

__device__ __forceinline__ float sigmoidf_(float x) { return 1.0f / (1.0f + expf(-x)); }

// ---------------------------------------------------------------------------
// LDS-tiled GEMM: block = 128 threads (4 waves), block tile = 64(M) x 64(N),
// wave w owns rows [64*bm + 16*w, +16), all 64 cols (4 WMMA accumulators).
// Both A and B K-panels (64x64) staged in LDS in a K-pair-swizzled layout:
//     lds[((k>>1)*64 + rc)*2 + (k&1)]
// so each lane's WMMA fragment is one contiguous ds_load_b64.
// Grid must be exactly (M/64)*(N/64): no divergence, EXEC all-1s for WMMA.
// Epilogue: optional bias[N], optional addend (stride ldc), optional relu.
// ---------------------------------------------------------------------------
__global__ void __launch_bounds__(128)
gemm_lds(const float* __restrict__ A, const float* __restrict__ B,
         float* __restrict__ C,
         const float* __restrict__ bias,
         const float* __restrict__ addend,
         int M, int N, int K, int lda, int ldb, int ldc,
         int relu_flag)
{
    __shared__ float ldsA[64 * 64];
    __shared__ float ldsB[64 * 64];

    int tid  = threadIdx.x;                 // 0..127
    int wave = tid >> 5, lane = tid & 31;
    int tilesN = N >> 6;
    int bm = blockIdx.x / tilesN, bn = blockIdx.x % tilesN;
    int row0 = bm << 6, col0 = bn << 6;
    int half = lane >> 4, l16 = lane & 15;
    int rowLocal = wave * 16 + l16;         // this lane's A row within block tile

    v8f acc0 = {}, acc1 = {}, acc2 = {}, acc3 = {};

    for (int kp = 0; kp < K; kp += 64) {
        __syncthreads();                    // previous-panel reads done
        // ---- stage A panel (64 rows x 64 k), swizzled by K pairs ----------
        for (int i = tid; i < 64 * 16; i += 128) {
            int r = i >> 4, c4 = i & 15;    // r = local row, c4*4 = local k
            float4 v = *(const float4*)(A + (size_t)(row0 + r) * lda + kp + c4 * 4);
            int base = (((c4 * 4) >> 1) * 64 + r) * 2;
            ldsA[base + 0]       = v.x;     // k, k+1 contiguous  -> ds_store_b64
            ldsA[base + 1]       = v.y;
            ldsA[base + 128 + 0] = v.z;     // next K pair group (+2*64)
            ldsA[base + 128 + 1] = v.w;
        }
        // ---- stage B panel (64 k x 64 cols), swizzled by K pairs ----------
        for (int i = tid; i < 64 * 16; i += 128) {
            int r = i >> 4, c4 = i & 15;    // r = local k, c4*4 = local col
            float4 v = *(const float4*)(B + (size_t)(kp + r) * ldb + col0 + c4 * 4);
            int base = ((r >> 1) * 64 + c4 * 4) * 2 + (r & 1);
            ldsB[base + 0] = v.x;
            ldsB[base + 2] = v.y;
            ldsB[base + 4] = v.z;
            ldsB[base + 6] = v.w;
        }
        // prefetch next panels into cache while we compute this one
        if (kp + 64 < K) {
            int r = tid & 63;
            __builtin_prefetch(A + (size_t)(row0 + r) * lda + kp + 64, 0, 0);
            __builtin_prefetch(B + (size_t)(kp + 64 + r) * ldb + col0, 0, 0);
        }
        __syncthreads();
        // ---- 16 k-steps of WMMA from LDS ---------------------------------
#pragma unroll
        for (int k = 0; k < 64; k += 4) {
            int kg = (k >> 1) + half;       // K-pair group for this half-wave
            v2f a = *(const v2f*)(ldsA + (kg * 64 + rowLocal) * 2);
            const float* bp = ldsB + (kg * 64 + l16) * 2;
            v2f b0 = *(const v2f*)(bp + 0);
            v2f b1 = *(const v2f*)(bp + 32);
            v2f b2 = *(const v2f*)(bp + 64);
            v2f b3 = *(const v2f*)(bp + 96);
            acc0 = wmma_f32(a, b0, acc0);
            acc1 = wmma_f32(a, b1, acc1);
            acc2 = wmma_f32(a, b2, acc2);
            acc3 = wmma_f32(a, b3, acc3);
        }
    }

    // ---- epilogue --------------------------------------------------------
    int rbase = row0 + wave * 16 + 8 * half;
#pragma unroll
    for (int r = 0; r < 8; ++r) {
        int row = rbase + r;
        float vals[4] = {acc0[r], acc1[r], acc2[r], acc3[r]};
#pragma unroll
        for (int s = 0; s < 4; ++s) {
            int col = col0 + s * 16 + l16;
            float v = vals[s];
            if (bias)      v += bias[col];
            if (addend)    v += addend[(size_t)row * ldc + col];
            if (relu_flag) v = fmaxf(v, 0.0f);
            C[(size_t)row * ldc + col] = v;
        }
    }
}

// ---------------------------------------------------------------------------
// GEMM with B transposed:  C[m][n] = sum_k A[m][k] * Bt[n][k]   (out @ out^T)
// B-transposed fragments are contiguous b64 loads already; register version.
// ---------------------------------------------------------------------------
__global__ void gemm16x64_bt(const float* __restrict__ A, const float* __restrict__ Bt,
                             float* __restrict__ C,
                             int M, int N, int K, int lda, int ldb, int ldc)
{
    int wave = (blockIdx.x * blockDim.x + threadIdx.x) >> 5;
    int lane = threadIdx.x & 31;
    int tilesN = N >> 6;
    int totalTiles = (M >> 4) * tilesN;
    if (wave >= totalTiles) return;         // wave-uniform
    int tm = wave / tilesN, tn = wave % tilesN;
    int row0 = tm << 4, col0 = tn << 6;
    int half = lane >> 4, l16 = lane & 15;

    v8f acc0 = {}, acc1 = {}, acc2 = {}, acc3 = {};
    const float* arow = A  + (size_t)(row0 + l16) * lda + 2 * half;
    const float* brow = Bt + (size_t)(col0 + l16) * ldb + 2 * half;

    for (int k = 0; k < K; k += 4) {
        v2f a  = *(const v2f*)(arow + k);
        v2f b0 = *(const v2f*)(brow + k);
        v2f b1 = *(const v2f*)(brow + 16 * (size_t)ldb + k);
        v2f b2 = *(const v2f*)(brow + 32 * (size_t)ldb + k);
        v2f b3 = *(const v2f*)(brow + 48 * (size_t)ldb + k);
        acc0 = wmma_f32(a, b0, acc0);
        acc1 = wmma_f32(a, b1, acc1);
        acc2 = wmma_f32(a, b2, acc2);
        acc3 = wmma_f32(a, b3, acc3);
    }

    int rbase = row0 + 8 * half;
#pragma unroll
    for (int r = 0; r < 8; ++r) {
        int row = rbase + r;
        float vals[4] = {acc0[r], acc1[r], acc2[r], acc3[r]};
#pragma unroll
        for (int s = 0; s < 4; ++s)
            C[(size_t)row * ldc + col0 + s * 16 + l16] = vals[s];
    }
}

// ---------------------------------------------------------------------------
// Geo precompute: Gs = coords @ geo_W1[0:4], Gd = coords @ geo_W1[4:8]  (K=4)
// ---------------------------------------------------------------------------
__global__ void geo_kernel(const float* __restrict__ coords, const float* __restrict__ geo_W1,
                           float* __restrict__ Gs, float* __restrict__ Gd)
{
    int idx = blockIdx.x * blockDim.x + threadIdx.x;
    if (idx >= 512 * 64) return;
    int n = idx >> 6, j = idx & 63;
    float c0 = coords[n * 4 + 0], c1 = coords[n * 4 + 1];
    float c2 = coords[n * 4 + 2], c3 = coords[n * 4 + 3];
    Gs[idx] = c0 * geo_W1[0 * 64 + j] + c1 * geo_W1[1 * 64 + j]
            + c2 * geo_W1[2 * 64 + j] + c3 * geo_W1[3 * 64 + j];
    Gd[idx] = c0 * geo_W1[4 * 64 + j] + c1 * geo_W1[5 * 64 + j]
            + c2 * geo_W1[6 * 64 + j] + c3 * geo_W1[7 * 64 + j];
}

// ---------------------------------------------------------------------------
// Per-pair edge kernel: both directions of (t, d). Writes edge weights already
// TRANSPOSED so the segment-sum becomes a plain row-major GEMM:
//   EW1T[d][t] = ew(edge t -> det d),  EW2T[t][d] = ew(edge det d -> t)
// Also computes IoU for the first-half edges.
// ---------------------------------------------------------------------------
__global__ void edge_kernel(const float* __restrict__ P,  const float* __restrict__ Q,
                            const float* __restrict__ Gs, const float* __restrict__ Gd,
                            const float* __restrict__ app_b1, const float* __restrict__ app_w2,
                            const float* __restrict__ app_b2,
                            const float* __restrict__ geo_b1, const float* __restrict__ geo_w2,
                            const float* __restrict__ geo_b2,
                            const float* __restrict__ aff_w, const float* __restrict__ aff_b,
                            const float* __restrict__ co,
                            float* __restrict__ EW1T, float* __restrict__ EW2T,
                            float* __restrict__ iouM)
{
    __shared__ float sP[128], sQ[128], sGs[64], sGd[64];
    int t = blockIdx.x;      // tracklet 0..255
    int d = threadIdx.x;     // det index 0..255
    int tid = threadIdx.x;
    if (tid < 128)                 sP[tid]        = P[t * 128 + tid];
    else                           sQ[tid - 128]  = Q[t * 128 + (tid - 128)];
    if (tid < 64)                  sGs[tid]       = Gs[t * 64 + tid];
    else if (tid < 128)            sGd[tid - 64]  = Gd[t * 64 + (tid - 64)];
    __syncthreads();

    int nd = 256 + d;
    const float* Pd = P + (size_t)nd * 128;
    const float* Qd = Q + (size_t)nd * 128;
    float s1 = 0.f, s1r = 0.f;
#pragma unroll 4
    for (int i = 0; i < 128; ++i) {
        float b1 = app_b1[i], w2 = app_w2[i];
        s1  += fmaxf(sP[i] + Qd[i] + b1, 0.f) * w2;   // edge t -> det
        s1r += fmaxf(Pd[i] + sQ[i] + b1, 0.f) * w2;   // edge det -> t
    }
    float x1  = sigmoidf_(s1  + app_b2[0]);
    float x1r = sigmoidf_(s1r + app_b2[0]);

    const float* Gsd = Gs + (size_t)nd * 64;
    const float* Gdd = Gd + (size_t)nd * 64;
    float s2 = 0.f, s2r = 0.f;
#pragma unroll 4
    for (int i = 0; i < 64; ++i) {
        float b1 = geo_b1[i], w2 = geo_w2[i];
        s2  += fmaxf(sGs[i] + Gdd[i] + b1, 0.f) * w2;
        s2r += fmaxf(Gsd[i] + sGd[i] + b1, 0.f) * w2;
    }
    float x2  = sigmoidf_(s2  + geo_b2[0]);
    float x2r = sigmoidf_(s2r + geo_b2[0]);

    float aw0 = aff_w[0], aw1 = aff_w[1], ab = aff_b[0];
    EW1T[d * 256 + t] = sigmoidf_(aw0 * x1  + aw1 * x2  + ab);
    EW2T[t * 256 + d] = sigmoidf_(aw0 * x1r + aw1 * x2r + ab);

    // IoU(co[t], co[256+d]) for first-half edges
    float ax0 = co[t * 4 + 0], ay0 = co[t * 4 + 1], ax1 = co[t * 4 + 2], ay1 = co[t * 4 + 3];
    float bx0 = co[nd * 4 + 0], by0 = co[nd * 4 + 1], bx1 = co[nd * 4 + 2], by1 = co[nd * 4 + 3];
    float ltx = fmaxf(ax0, bx0), lty = fmaxf(ay0, by0);
    float rbx = fminf(ax1, bx1), rby = fminf(ay1, by1);
    float w   = fmaxf(rbx - ltx, 0.f), hgt = fmaxf(rby - lty, 0.f);
    float inter = w * hgt;
    float areaA = (ax1 - ax0) * (ay1 - ay0);
    float areaB = (bx1 - bx0) * (by1 - by0);
    iouM[t * 256 + d] = inter / (areaA + areaB - inter + 1e-6f);
}

// ---------------------------------------------------------------------------
// Row norms of out (512x512), clipped at 1e-6.
// ---------------------------------------------------------------------------
__global__ void norms_kernel(const float* __restrict__ outM, float* __restrict__ norms)
{
    int n = blockIdx.x * blockDim.x + threadIdx.x;
    if (n >= 512) return;
    const float* r = outM + (size_t)n * 512;
    float s = 0.f;
    for (int k = 0; k < 512; ++k) s += r[k] * r[k];
    norms[n] = fmaxf(sqrtf(s), 1e-6f);
}

// ---------------------------------------------------------------------------
// Build the (T+1)x(D+1) = 257x257 Sinkhorn matrix.
// ---------------------------------------------------------------------------
__global__ void build_M(const float* __restrict__ S, const float* __restrict__ norms,
                        const float* __restrict__ iouM,
                        const float* __restrict__ fin_w, const float* __restrict__ fin_b,
                        float* __restrict__ M)
{
    int idx = blockIdx.x * blockDim.x + threadIdx.x;
    if (idx >= 257 * 257) return;
    int t = idx / 257, d = idx % 257;
    float sv = expf(-1.0f);                         // exp(SLACK * LAM)
    float v;
    if (t == 256 || d == 256) {
        v = sv;
    } else {
        float cosv = S[t * 256 + d] / (norms[t] * norms[256 + d]);
        float fin  = sigmoidf_(fin_w[0] * cosv + fin_w[1] * iouM[t * 256 + d] + fin_b[0]);
        v = expf(5.0f * fin);                       // LAM = 5
    }
    M[idx] = v;
}

// ---------------------------------------------------------------------------
// Sinkhorn normalization steps (row then column), 257x257, one block per line.
// ---------------------------------------------------------------------------
__global__ void sink_row(float* __restrict__ M)
{
    __shared__ float red[256];
    int row = blockIdx.x;
    float* mrow = M + (size_t)row * 257;
    float s = 0.f;
    for (int c = threadIdx.x; c < 257; c += 256) s += mrow[c];
    red[threadIdx.x] = s;
    __syncthreads();
    for (int off = 128; off > 0; off >>= 1) {
        if (threadIdx.x < off) red[threadIdx.x] += red[threadIdx.x + off];
        __syncthreads();
    }
    float inv = 1.0f / red[0];
    for (int c = threadIdx.x; c < 257; c += 256) mrow[c] *= inv;
}

__global__ void sink_col(float* __restrict__ M)
{
    __shared__ float red[256];
    int col = blockIdx.x;
    float s = 0.f;
    for (int r = threadIdx.x; r < 257; r += 256) s += M[(size_t)r * 257 + col];
    red[threadIdx.x] = s;
    __syncthreads();
    for (int off = 128; off > 0; off >>= 1) {
        if (threadIdx.x < off) red[threadIdx.x] += red[threadIdx.x + off];
        __syncthreads();
    }
    float inv = 1.0f / red[0];
    for (int r = threadIdx.x; r < 257; r += 256) M[(size_t)r * 257 + col] *= inv;
}

// ---------------------------------------------------------------------------
// Final output: [ M[:256,:256].flat (65536) | ground_truth (65536) | D | T ]
// det_num / tracklet_num are int32 in the reference -> write raw int bits.
// ---------------------------------------------------------------------------
__global__ void finalize_kernel(const float* __restrict__ M, const float* __restrict__ gt,
                                float* __restrict__ out)
{
    int idx = blockIdx.x * blockDim.x + threadIdx.x;
    if (idx < 65536) {
        int t = idx >> 8, d = idx & 255;
        out[idx] = M[t * 257 + d];
    } else if (idx < 131072) {
        out[idx] = gt[idx - 65536];
    } else if (idx == 131072) {
        ((int*)out)[131072] = 256;      // det_num = D
    } else if (idx == 131073) {
        ((int*)out)[131073] = 256;      // tracklet_num = T
    }
}

// ---------------------------------------------------------------------------
extern "C" void kernel_launch(void* const* d_in, const int* in_sizes, int n_in,
                              void* d_out, int out_size, void* d_ws, size_t ws_size,
                              hipStream_t stream)
{
    (void)in_sizes; (void)n_in; (void)out_size; (void)ws_size;

    const float* x       = (const float*)d_in[0];   // 512 x 2048
    const float* coords  = (const float*)d_in[1];   // 512 x 4
    const float* co      = (const float*)d_in[2];   // 512 x 4 (coords_original)
    const float* gt      = (const float*)d_in[3];   // 65536
    /* d_in[4] = edge_index, structure is known (complete bipartite), unused */
    const float* W_enc   = (const float*)d_in[5];   // 2048 x 512
    const float* b_enc   = (const float*)d_in[6];   // 512
    const float* app_W1  = (const float*)d_in[7];   // 1024 x 128
    const float* app_b1  = (const float*)d_in[8];   // 128
    const float* app_w2  = (const float*)d_in[9];   // 128
    const float* app_b2  = (const float*)d_in[10];  // 1
    const float* geo_W1  = (const float*)d_in[11];  // 8 x 64
    const float* geo_b1  = (const float*)d_in[12];  // 64
    const float* geo_w2  = (const float*)d_in[13];  // 64
    const float* geo_b2  = (const float*)d_in[14];  // 1
    const float* aff_w   = (const float*)d_in[15];  // 2
    const float* aff_b   = (const float*)d_in[16];  // 1
    const float* W_self  = (const float*)d_in[17];  // 512 x 512
    const float* W_nbr   = (const float*)d_in[18];  // 512 x 512
    const float* b_mp    = (const float*)d_in[19];  // 512
    const float* fin_w   = (const float*)d_in[20];  // 2
    const float* fin_b   = (const float*)d_in[21];  // 1

    float* ws = (float*)d_ws;
    float* h     = ws;  ws += 512 * 512;
    float* Pm    = ws;  ws += 512 * 128;
    float* Qm    = ws;  ws += 512 * 128;
    float* hn    = ws;  ws += 512 * 512;
    float* hself = ws;  ws += 512 * 512;
    float* Gs    = ws;  ws += 512 * 64;
    float* Gd    = ws;  ws += 512 * 64;
    float* EW1T  = ws;  ws += 256 * 256;
    float* EW2T  = ws;  ws += 256 * 256;
    float* iouM  = ws;  ws += 256 * 256;
    float* outM  = ws;  ws += 512 * 512;
    float* Sm    = ws;  ws += 256 * 256;
    float* norms = ws;  ws += 512;
    float* Mm    = ws;  ws += 257 * 257;

    dim3 blk(128);  // 4 waves/block; grids sized exactly -> no divergence

    // 1. encoder: h = relu(x @ W_enc + b_enc)        [512,2048]x[2048,512]
    gemm_lds<<<64, blk, 0, stream>>>(x, W_enc, h, b_enc, nullptr,
                                     512, 512, 2048, 2048, 512, 512, 1);
    // 2. edge-MLP split: P = h @ W1_top, Q = h @ W1_bot   [512,512]x[512,128]
    gemm_lds<<<16, blk, 0, stream>>>(h, app_W1,             Pm, nullptr, nullptr,
                                     512, 128, 512, 512, 128, 128, 0);
    gemm_lds<<<16, blk, 0, stream>>>(h, app_W1 + 512 * 128, Qm, nullptr, nullptr,
                                     512, 128, 512, 512, 128, 128, 0);
    // 3. message precompute: hn = h @ W_nbr, hself = h @ W_self
    gemm_lds<<<64, blk, 0, stream>>>(h, W_nbr,  hn,    nullptr, nullptr,
                                     512, 512, 512, 512, 512, 512, 0);
    gemm_lds<<<64, blk, 0, stream>>>(h, W_self, hself, nullptr, nullptr,
                                     512, 512, 512, 512, 512, 512, 0);
    // 4. geo precompute
    geo_kernel<<<(512 * 64) / 256, 256, 0, stream>>>(coords, geo_W1, Gs, Gd);
    // 5. all 131072 edge weights (both directions) + IoU
    edge_kernel<<<256, 256, 0, stream>>>(Pm, Qm, Gs, Gd, app_b1, app_w2, app_b2,
                                         geo_b1, geo_w2, geo_b2, aff_w, aff_b, co,
                                         EW1T, EW2T, iouM);
    // 6. segment-sum as GEMM, fused with relu(hself + agg + b_mp):
    //    out[det rows] = relu(EW1T @ hn[0:256]   + hself[256:] + b_mp)
    //    out[trk rows] = relu(EW2T @ hn[256:512] + hself[0:256] + b_mp)
    gemm_lds<<<32, blk, 0, stream>>>(EW1T, hn,             outM + 256 * 512, b_mp,
                                     hself + 256 * 512, 256, 512, 256, 256, 512, 512, 1);
    gemm_lds<<<32, blk, 0, stream>>>(EW2T, hn + 256 * 512, outM,             b_mp,
                                     hself,             256, 512, 256, 256, 512, 512, 1);
    // 7. cosine: norms + S = out_trk @ out_det^T
    norms_kernel<<<2, 256, 0, stream>>>(outM, norms);
    gemm16x64_bt<<<16, blk, 0, stream>>>(outM, outM + 256 * 512, Sm,
                                         256, 256, 512, 512, 512, 256);
    // 8. Sinkhorn matrix + 8 normalization iterations
    build_M<<<(257 * 257 + 255) / 256, 256, 0, stream>>>(Sm, norms, iouM, fin_w, fin_b, Mm);
    for (int it = 0; it < 8; ++it) {
        sink_row<<<257, 256, 0, stream>>>(Mm);
        sink_col<<<257, 256, 0, stream>>>(Mm);
    }
    // 9. outputs
    finalize_kernel<<<(131074 + 255) / 256, 256, 0, stream>>>(Mm, gt, (float*)d_out);
}